// DSA_43722767073506
// MI455X (gfx1250) — compile-verified
//
#include <hip/hip_runtime.h>
#include <math.h>

// DSA reference collapses: log_softmax over size-1 axis == 0 exactly, so the
// attention context is identically zero and the 99-step scan reduces to ONE
// decoder-LSTM step on d[:, T-2]. Encoder / s1 / att_* are dead. Remaining
// live work: 32x128 LSTM pointwise + (32x128)x(128x128) GEMM + (1x128) dot.
// Single fused workgroup; GEMM runs on v_wmma_f32_16x16x4_f32.

typedef float v2f __attribute__((ext_vector_type(2)));
typedef float v8f __attribute__((ext_vector_type(8)));

#define B_ 32
#define T_ 100
#define H_ 128

__global__ __launch_bounds__(512) void dsa_fused_kernel(
    const float* __restrict__ d,       // (B, T-1)
    const float* __restrict__ W_ih_d,  // (4H, 1)
    const float* __restrict__ b_d,     // (4H,)
    const float* __restrict__ dec_w,   // (1, H+1)  -> only [0] matters (ctx==0)
    const float* __restrict__ dec_b,   // (1,)
    const float* __restrict__ d1_w,    // (H, 2H)   -> only [:, :H] matters
    const float* __restrict__ d1_b,    // (H,)
    const float* __restrict__ d2_w,    // (1, H)
    const float* __restrict__ d2_b,    // (1,)
    float* __restrict__ out)           // (B,)
{
    __shared__ float lds_h[B_ * H_];        // h_s : 32x128 (16 KB)
    __shared__ float lds_p[16 * 8 * 32];    // per-wave scaled partials (16 KB)

    const int tid = threadIdx.x;

    // ---------------- Phase 0: single decoder-LSTM step (zero h0,c0) -------
    // h_s[b,h] = sigmoid(zo)*tanh( sigmoid(zi)*tanh(zg) ),  z* = din*W + b
    const float dw0 = dec_w[0];
    const float db0 = dec_b[0];
    for (int idx = tid; idx < B_ * H_; idx += 512) {
        const int b = idx >> 7;          // /H_
        const int h = idx & (H_ - 1);
        const float din = d[b * (T_ - 1) + (T_ - 2)] * dw0 + db0;
        const float zi = din * W_ih_d[h]          + b_d[h];
        const float zg = din * W_ih_d[2 * H_ + h] + b_d[2 * H_ + h];
        const float zo = din * W_ih_d[3 * H_ + h] + b_d[3 * H_ + h];
        const float c  = (1.0f / (1.0f + expf(-zi))) * tanhf(zg);
        lds_h[idx]     = (1.0f / (1.0f + expf(-zo))) * tanhf(c);
    }
    __syncthreads();

    // ---------------- Phase 1: hid = h_s(32x128) x d1_w[:, :H]^T (128x128) -
    // 16 waves: wave = m_tile*8 + n_tile; each owns a 16x16 output tile and
    // accumulates 32 x V_WMMA_F32_16X16X4_F32 over K=128.
    const int wave   = tid >> 5;
    const int lane   = tid & 31;
    const int m_tile = wave >> 3;             // 0..1  (batch rows)
    const int n_tile = wave & 7;              // 0..7  (hidden cols)
    const int col    = lane & 15;
    const int koff   = (lane >> 4) << 1;      // lanes 0-15 -> K 0,1 ; 16-31 -> K 2,3
    const int rowA   = m_tile * 16 + col;     // A: M = batch row
    const int colB   = n_tile * 16 + col;     // B: N = output hidden index

    const float* __restrict__ aRow = lds_h + rowA * H_;          // LDS, stride H
    const float* __restrict__ bRow = d1_w  + colB * (2 * H_);    // global, row-major

    v8f acc = {0.f, 0.f, 0.f, 0.f, 0.f, 0.f, 0.f, 0.f};
#pragma unroll 4
    for (int k4 = 0; k4 < H_; k4 += 4) {
        v2f a, bf;
        a.x  = aRow[k4 + koff];
        a.y  = aRow[k4 + koff + 1];
        bf.x = bRow[k4 + koff];
        bf.y = bRow[k4 + koff + 1];
        // 8 args: (neg_a, A, neg_b, B, c_mod, C, reuse_a, reuse_b)
        acc = __builtin_amdgcn_wmma_f32_16x16x4_f32(
            false, a, false, bf, (short)0, acc, false, false);
    }

    // Scale each accumulator element by its d2_w column weight and stash.
    // D layout: lanes 0-15: D[r, col]; lanes 16-31: D[8+r, col].
    const float scale = d2_w[colB];
    {
        float* p = lds_p + wave * 256 + lane;
#pragma unroll
        for (int r = 0; r < 8; ++r) p[r * 32] = scale * acc[r];
    }
    __syncthreads();

    // ---------------- Phase 2: deterministic fixed-order reduction ---------
    // out[b] = sum_h d2_w[h]*hid[b,h] + (d2_w . d1_b) + d2_b   (ctx half == 0)
    if (tid < B_) {
        const int mt = tid >> 4;
        const int M  = tid & 15;
        const int r  = (M < 8) ? M : (M - 8);
        const int l0 = (M < 8) ? 0 : 16;
        float sum = 0.f;
        for (int n = 0; n < 8; ++n) {
            const float* q = lds_p + (mt * 8 + n) * 256 + r * 32 + l0;
#pragma unroll
            for (int l = 0; l < 16; ++l) sum += q[l];
        }
        float c0 = d2_b[0];
        for (int h = 0; h < H_; ++h) c0 += d2_w[h] * d1_b[h];
        out[tid] = sum + c0;
    }
}

extern "C" void kernel_launch(void* const* d_in, const int* in_sizes, int n_in,
                              void* d_out, int out_size, void* d_ws, size_t ws_size,
                              hipStream_t stream) {
    (void)in_sizes; (void)n_in; (void)out_size; (void)d_ws; (void)ws_size;
    // setup_inputs order:
    // 0:x 1:d 2:W_ih_e 3:b_e 4:W_ih_d 5:b_d 6:att_w1 7:att_b1 8:att_w2
    // 9:att_b2 10:att_v 11:att_vb 12:dec_w 13:dec_b 14:d1_w 15:d1_b
    // 16:d2_w 17:d2_b       (0,2,3,6..11 are mathematically dead)
    const float* d_    = (const float*)d_in[1];
    const float* Wihd  = (const float*)d_in[4];
    const float* bd    = (const float*)d_in[5];
    const float* decw  = (const float*)d_in[12];
    const float* decb  = (const float*)d_in[13];
    const float* d1w   = (const float*)d_in[14];
    const float* d1b   = (const float*)d_in[15];
    const float* d2w   = (const float*)d_in[16];
    const float* d2b   = (const float*)d_in[17];
    float* out = (float*)d_out;

    dsa_fused_kernel<<<1, 512, 0, stream>>>(d_, Wihd, bd, decw, decb,
                                            d1w, d1b, d2w, d2b, out);
}